// PartSampler_34892314313151
// MI455X (gfx1250) — compile-verified
//
#include <hip/hip_runtime.h>
#include <hip/hip_bf16.h>
#include <math.h>

// PartSampler for MI455X (gfx1250, wave32).
// feat: (16,1024,64,64) f32 = 256MB. Two mandatory passes over feat ->
// ~512MB @ 23.3TB/s ~= 22us floor. Pass 2 (channel contraction) is done with
// V_WMMA_F32_16X16X4_F32: two accumulators (ones-vector and a-vector) share
// one B-matrix load of feat, so matrix ops ride along at streaming bandwidth.

typedef __attribute__((ext_vector_type(2))) float v2f;
typedef __attribute__((ext_vector_type(8))) float v8f;

#define BATCH 16
#define CCH   1024
#define HH    64
#define WW    64
#define HW    4096
#define KPK   8
#define NMSR  6      // int(0.1 * 64)
#define RHALF 2      // R//2, R=5

// ---------------- Kernel 1: chanSum[b,c] = sum_hw feat  (pass 1, 256MB) ----
__global__ void k_chan_sum(const float* __restrict__ feat,
                           float* __restrict__ chanSum) {
    const int c = blockIdx.x, b = blockIdx.y, t = threadIdx.x;
    const float4* plane = (const float4*)(feat + ((size_t)b * CCH + c) * HW);
    float s = 0.f;
#pragma unroll
    for (int j = 0; j < 4; ++j) {                 // 4096 floats = 1024 float4
        float4 v = plane[t + 256 * j];
        s += v.x + v.y + v.z + v.w;
    }
    __shared__ float red[256];
    red[t] = s;
    __syncthreads();
    for (int off = 128; off > 0; off >>= 1) {
        if (t < off) red[t] += red[t + off];
        __syncthreads();
    }
    if (t == 0) chanSum[b * CCH + c] = red[0];
}

// ---------------- Kernel 2: a[b,:] = softmax_c(chanSum/HW) -----------------
__global__ void k_softmax_a(const float* __restrict__ chanSum,
                            float* __restrict__ a) {
    const int b = blockIdx.x, t = threadIdx.x;    // 1024 threads
    __shared__ float red[1024];
    const float m = chanSum[b * CCH + t] * (1.0f / HW);
    red[t] = m;
    __syncthreads();
    for (int off = 512; off > 0; off >>= 1) {
        if (t < off) red[t] = fmaxf(red[t], red[t + off]);
        __syncthreads();
    }
    const float mx = red[0];
    __syncthreads();
    const float e = expf(m - mx);
    red[t] = e;
    __syncthreads();
    for (int off = 512; off > 0; off >>= 1) {
        if (t < off) red[t] += red[t + off];
        __syncthreads();
    }
    a[b * CCH + t] = e / red[0];
}

// ---------------- Kernel 3 (WMMA): pass 2 over feat ------------------------
// Per wave: 16 pixels. Loop c0 in steps of 4 channels.
//   B (4x16 f32, 2 VGPR): v0 = F[c0+csel, p], v1 = F[c0+csel+1, p]
//     with csel = 0 (lanes 0-15) / 2 (lanes 16-31), p = pixel per lane%16.
//   A_ones -> accS = sum_c F           (for spatial softmax)
//   A_a    -> accW = sum_c a_c F       (saliency numerator)
// A 16x4 layout: v0 = K0|K2, v1 = K1|K3, all rows identical -> every D row
// equals the result; read acc[0] from lanes 0-15 (D: VGPR0 = M0,N=lane).
__global__ void k_spat_wsum(const float* __restrict__ feat,
                            const float* __restrict__ a,
                            float* __restrict__ spat,
                            float* __restrict__ wsum) {
    const int b    = blockIdx.y;
    const int tid  = threadIdx.x;
    const int lane = tid & 31;
    const int wv   = tid >> 5;                    // 8 waves / block
    const int hw0  = blockIdx.x * 128 + wv * 16;  // 32 blocks * 128 px = 4096

    __shared__ float aL[CCH];
    for (int i = tid; i < CCH; i += 256) aL[i] = a[b * CCH + i];
    __syncthreads();

    const float* fb   = feat + (size_t)b * CCH * HW;
    const int    p    = hw0 + (lane & 15);
    const int    csel = (lane >> 4) << 1;         // 0 or 2

    v8f accS = {};
    v8f accW = {};
    v2f As; As[0] = 1.0f; As[1] = 1.0f;

    for (int c0 = 0; c0 < CCH; c0 += 4) {
        v2f Aw; Aw[0] = aL[c0 + csel]; Aw[1] = aL[c0 + csel + 1];
        const float* fp = fb + (size_t)(c0 + csel) * HW + p;
        v2f Bv; Bv[0] = fp[0]; Bv[1] = fp[HW];
        if (c0 + 4 < CCH) __builtin_prefetch(fp + 4 * HW, 0, 1);
        accS = __builtin_amdgcn_wmma_f32_16x16x4_f32(
            false, As, false, Bv, (short)0, accS, false, false);
        accW = __builtin_amdgcn_wmma_f32_16x16x4_f32(
            false, Aw, false, Bv, (short)0, accW, false, false);
    }
    if (lane < 16) {
        spat[b * HW + hw0 + lane] = accS[0];
        wsum[b * HW + hw0 + lane] = accW[0];
    }
}

// ---------------- Kernel 4: S = softmax_hw(spat*w/C); M = wsum*S -----------
__global__ void k_softmax_S_M(const float* __restrict__ spat,
                              const float* __restrict__ wsum,
                              const float* __restrict__ spatial_w,
                              float* __restrict__ M) {
    const int b = blockIdx.x, t = threadIdx.x;    // 256 threads, 16 px each
    const float wsc = spatial_w[0] * (1.0f / CCH);
    float l[16];
    float mx = -INFINITY;
#pragma unroll
    for (int j = 0; j < 16; ++j) {
        l[j] = spat[b * HW + t + 256 * j] * wsc;
        mx = fmaxf(mx, l[j]);
    }
    __shared__ float red[256];
    red[t] = mx;
    __syncthreads();
    for (int off = 128; off > 0; off >>= 1) {
        if (t < off) red[t] = fmaxf(red[t], red[t + off]);
        __syncthreads();
    }
    mx = red[0];
    __syncthreads();
    float se = 0.f;
#pragma unroll
    for (int j = 0; j < 16; ++j) { l[j] = expf(l[j] - mx); se += l[j]; }
    red[t] = se;
    __syncthreads();
    for (int off = 128; off > 0; off >>= 1) {
        if (t < off) red[t] += red[t + off];
        __syncthreads();
    }
    const float inv = 1.0f / red[0];
#pragma unroll
    for (int j = 0; j < 16; ++j) {
        const int i = t + 256 * j;
        M[b * HW + i] = wsum[b * HW + i] * l[j] * inv;
    }
}

// ---------------- Kernel 5: iterative argmax + NMS in LDS ------------------
__global__ void k_nms(const float* __restrict__ M,
                      int* __restrict__ peaksI,
                      float* __restrict__ outPeaks) {
    const int b = blockIdx.x, t = threadIdx.x;    // 256 threads
    __shared__ float Mv[HW];
    __shared__ float rv[256];
    __shared__ int   ri[256];
    __shared__ int   sh, sw;
#pragma unroll
    for (int j = 0; j < 16; ++j) Mv[t + 256 * j] = M[b * HW + t + 256 * j];
    __syncthreads();

    for (int k = 0; k < KPK; ++k) {
        // per-thread argmax with first-flat-index tie-break (jnp.argmax)
        float bv = -INFINITY;
        int   bi = 0x7fffffff;
#pragma unroll
        for (int j = 0; j < 16; ++j) {
            const int i = t + 256 * j;
            const float v = Mv[i];
            if (v > bv || (v == bv && i < bi)) { bv = v; bi = i; }
        }
        rv[t] = bv; ri[t] = bi;
        __syncthreads();
        for (int off = 128; off > 0; off >>= 1) {
            if (t < off) {
                const float v2 = rv[t + off]; const int i2 = ri[t + off];
                if (v2 > rv[t] || (v2 == rv[t] && i2 < ri[t])) {
                    rv[t] = v2; ri[t] = i2;
                }
            }
            __syncthreads();
        }
        if (t == 0) {
            const int idx = ri[0];
            const int h = idx >> 6, w = idx & 63;
            sh = h; sw = w;
            peaksI[(b * KPK + k) * 2 + 0] = h;
            peaksI[(b * KPK + k) * 2 + 1] = w;
            outPeaks[(b * KPK + k) * 2 + 0] = (float)h;
            outPeaks[(b * KPK + k) * 2 + 1] = (float)w;
        }
        __syncthreads();
        const int h = sh, w = sw;
#pragma unroll
        for (int j = 0; j < 16; ++j) {
            const int i = t + 256 * j;
            const int r = i >> 6, c = i & 63;
            if (r >= h - NMSR && r <= h + NMSR &&
                c >= w - NMSR && c <= w + NMSR)
                Mv[i] = -INFINITY;
        }
        __syncthreads();
    }
}

// ---------------- Kernel 6: Z[b,k,c] = mean of clamped 5x5 window ----------
__global__ void k_parts(const float* __restrict__ feat,
                        const int* __restrict__ peaksI,
                        float* __restrict__ Z) {
    const int k = blockIdx.x, b = blockIdx.y, t = threadIdx.x;
    const int h = peaksI[(b * KPK + k) * 2 + 0];
    const int w = peaksI[(b * KPK + k) * 2 + 1];
    const int h0 = max(0, h - RHALF), h1 = min(HH, h + RHALF + 1);
    const int w0 = max(0, w - RHALF), w1 = min(WW, w + RHALF + 1);
    const float inva = 1.0f / (float)((h1 - h0) * (w1 - w0));
    for (int c = t; c < CCH; c += 256) {
        const float* pl = feat + ((size_t)b * CCH + c) * HW;
        float s = 0.f;
        for (int r = h0; r < h1; ++r)
            for (int cc = w0; cc < w1; ++cc)
                s += pl[r * WW + cc];
        Z[(b * KPK + k) * CCH + c] = s * inva;
    }
}

extern "C" void kernel_launch(void* const* d_in, const int* in_sizes, int n_in,
                              void* d_out, int out_size, void* d_ws, size_t ws_size,
                              hipStream_t stream) {
    const float* feat      = (const float*)d_in[0];
    const float* spatial_w = (const float*)d_in[1];
    float* out = (float*)d_out;

    // workspace layout (floats): ~918KB total
    float* ws       = (float*)d_ws;
    float* chanSum  = ws;                 // 16*1024
    float* a        = chanSum + BATCH * CCH;
    float* spat     = a + BATCH * CCH;    // 16*4096
    float* wsum     = spat + BATCH * HW;
    float* M        = wsum + BATCH * HW;
    int*   peaksI   = (int*)(M + BATCH * HW);   // 16*8*2 ints

    float* Z        = out;                        // (16,8,1024)
    float* outPeaks = out + BATCH * KPK * CCH;    // (16,8,2) as f32

    k_chan_sum   <<<dim3(CCH, BATCH), 256,  0, stream>>>(feat, chanSum);
    k_softmax_a  <<<dim3(BATCH),      1024, 0, stream>>>(chanSum, a);
    k_spat_wsum  <<<dim3(32, BATCH),  256,  0, stream>>>(feat, a, spat, wsum);
    k_softmax_S_M<<<dim3(BATCH),      256,  0, stream>>>(spat, wsum, spatial_w, M);
    k_nms        <<<dim3(BATCH),      256,  0, stream>>>(M, peaksI, outPeaks);
    k_parts      <<<dim3(KPK, BATCH), 256,  0, stream>>>(feat, peaksI, Z);
}